// AdaPool3d_309237645648
// MI455X (gfx1250) — compile-verified
//
#include <hip/hip_runtime.h>
#include <math.h>

// AdaPool3d (K=2): x [4,64,16,112,112] f32, beta [8,56,56] f32 -> out [4,64,8,56,56] f32
// Block = one (b*c, od) slab of 56x28 output pairs; 224 threads (7 wave32) x 7 iters.
// All per-iteration addressing is constant pointer increments (no div/mod in loop).

#define LOG2E     1.4426950408889634f
#define FLT_MAX_V 3.4028234663852886e38f

#define DSTR 12544           // 112*112 : x stride for d
#define HSTR 112             // x stride for h
#define XSTEP (8 * 2 * HSTR) // 1792 : x advance per iteration (8 oh rows)
#define OSTEP (8 * 56)       // 448  : out/beta advance per iteration

// One output element from its 8-element window + beta.
__device__ __forceinline__ float adapool_one(const float v[8], float beta) {
    // window mean (balanced tree)
    const float s01 = v[0] + v[1], s23 = v[2] + v[3];
    const float s45 = v[4] + v[5], s67 = v[6] + v[7];
    const float a  = ((s01 + s23) + (s45 + s67)) * 0.125f;
    const float a2 = a * a;
    const float c  = (2.0f * LOG2E) * a;   // folded dsc numerator scale

    // max for the value-softmax (balanced tree); dsc-softmax needs no max:
    // |2av/(a^2+v^2)| <= 1 so exp2(dsc*log2e) is in [0.37, 2.72].
    const float m01 = fmaxf(v[0], v[1]), m23 = fmaxf(v[2], v[3]);
    const float m45 = fmaxf(v[4], v[5]), m67 = fmaxf(v[6], v[7]);
    const float mb  = fmaxf(fmaxf(m01, m23), fmaxf(m45, m67)) * LOG2E;

    float den_e = 0.0f, num_e = 0.0f;   // SoftPool (exp-max) softmax
    float den_d = 0.0f, num_d = 0.0f;   // EDSCW softmax
#pragma unroll
    for (int i = 0; i < 8; ++i) {
        const float vi = v[i];
        // exp(vi - m) == exp2(vi*log2e - m*log2e)
        const float e = __builtin_amdgcn_exp2f(__builtin_fmaf(vi, LOG2E, -mb));
        den_e += e;
        num_e  = __builtin_fmaf(e, vi, num_e);

        const float denom = __builtin_fmaf(vi, vi, a2);
        const float r     = __builtin_amdgcn_rcpf(denom);
        const float u     = c * vi;
        // dsc pre-scaled by log2e so exp2 gives exp(dsc); where(denom>0, ..., 0)
        const float dscs  = (denom > 0.0f) ? u * r : 0.0f;
        const float ed    = __builtin_amdgcn_exp2f(dscs);
        den_d += ed;
        num_d  = __builtin_fmaf(ed, vi, num_d);
    }
    const float em    = num_e * __builtin_amdgcn_rcpf(den_e);
    const float edscw = num_d * __builtin_amdgcn_rcpf(den_d);

    float out = __builtin_fmaf(beta, edscw - em, em);   // beta*edscw + (1-beta)*em
    // nan_to_num: nan->0, +/-inf -> +/-FLT_MAX
    out = (out != out) ? 0.0f : fminf(fmaxf(out, -FLT_MAX_V), FLT_MAX_V);
    return out;
}

__global__ __launch_bounds__(224) void adapool3d_kernel(
    const float* __restrict__ x,
    const float* __restrict__ beta,
    float* __restrict__ out)
{
    // One slab per block: bc in [0,256), od in [0,8)
    const int bc = blockIdx.x >> 3;
    const int od = blockIdx.x & 7;

    // One-time per-thread decomposition (only div/mod in the kernel)
    const int tid = threadIdx.x;
    const int owp = tid % 28;     // ow pair 0..27
    const int oh0 = tid / 28;     // 0..7

    const float* xp = x    + (bc * 16 + 2 * od) * DSTR + (2 * oh0) * HSTR + 4 * owp;
    const float* bp = beta + (od * 56 + oh0) * 56 + 2 * owp;
    float*       op = out  + ((bc * 8 + od) * 56 + oh0) * 56 + 2 * owp;

#pragma unroll 1
    for (int it = 0; it < 7; ++it) {
        // Prefetch next iteration's four cache lines (global_prefetch_b8 with
        // immediate offsets off the same address register) to overlap HBM
        // latency with the 26-deep transcendental chain below.
        if (it != 6) {
            __builtin_prefetch(xp + XSTEP,               0, 3);
            __builtin_prefetch(xp + XSTEP + HSTR,        0, 3);
            __builtin_prefetch(xp + XSTEP + DSTR,        0, 3);
            __builtin_prefetch(xp + XSTEP + DSTR + HSTR, 0, 3);
        }

        // Four window rows, float4 each (16B-aligned)
        const float4 r00 = *reinterpret_cast<const float4*>(xp);
        const float4 r01 = *reinterpret_cast<const float4*>(xp + HSTR);
        const float4 r10 = *reinterpret_cast<const float4*>(xp + DSTR);
        const float4 r11 = *reinterpret_cast<const float4*>(xp + DSTR + HSTR);

        const float2 b2 = *reinterpret_cast<const float2*>(bp);

        const float v0[8] = { r00.x, r00.y, r01.x, r01.y, r10.x, r10.y, r11.x, r11.y };
        const float v1[8] = { r00.z, r00.w, r01.z, r01.w, r10.z, r10.w, r11.z, r11.w };

        float2 o;
        o.x = adapool_one(v0, b2.x);
        o.y = adapool_one(v1, b2.y);
        *reinterpret_cast<float2*>(op) = o;

        xp += XSTEP;
        bp += OSTEP;
        op += OSTEP;
    }
}

extern "C" void kernel_launch(void* const* d_in, const int* in_sizes, int n_in,
                              void* d_out, int out_size, void* d_ws, size_t ws_size,
                              hipStream_t stream) {
    (void)in_sizes; (void)n_in; (void)d_ws; (void)ws_size; (void)out_size;
    const float* x    = (const float*)d_in[0];
    const float* beta = (const float*)d_in[1];
    float* out        = (float*)d_out;

    // 256 (b*c) * 8 (od) slabs, one per block; 224 threads = 7 wave32s.
    adapool3d_kernel<<<2048, 224, 0, stream>>>(x, beta, out);
}